// MultiBoxLoss_53403623358616
// MI455X (gfx1250) — compile-verified
//
#include <hip/hip_runtime.h>
#include <stdint.h>

#define BDIM   64
#define DDIM   65536
#define NEGPOS 3

// ---- workspace layout (indexed as int/float, same buffer) ----
#define WS_NUMPOS  0     // 64 ints   : per-row positive count (atomic int)
#define WS_POSBCE  64    // 64 floats : per-row sum of bce over positives
#define WS_NEGSEL  128   // 64 floats : per-row sum of bce over selected negatives
#define WS_LOCPART 192   // 4096 floats : per-block smooth-L1 partials
#define LOC_BLOCKS_PER_ROW 64
#define NPART (BDIM * LOC_BLOCKS_PER_ROW)

// conf_kernel dynamic LDS: row values + 2048-bin histogram + scratch
#define SMEM_BYTES (DDIM * 4 + 2048 * 4 + 64)

__global__ void init_ws_kernel(int* __restrict__ wsI) {
  int t = threadIdx.x;
  if (t < BDIM) wsI[WS_NUMPOS + t] = 0;
}

__device__ __forceinline__ float smooth_l1(float d) {
  float ad = fabsf(d);
  return (ad < 1.0f) ? 0.5f * d * d : ad - 0.5f;
}

// Pass 1: masked smooth-L1 (only touch loc data where conf_t > 0) + per-row num_pos.
__global__ __launch_bounds__(256) void loc_kernel(
    const float* __restrict__ loc_data, const float* __restrict__ loc_t,
    const int* __restrict__ conf_t, int* __restrict__ wsI, float* __restrict__ wsF) {
  int b = blockIdx.y;
  int chunk = blockIdx.x;
  int base = chunk * (256 * 4) + threadIdx.x * 4;  // element index within row
  size_t rowoff = (size_t)b * DDIM;

  const int4 ct4 = *(const int4*)(conf_t + rowoff + base);
  int cts[4] = {ct4.x, ct4.y, ct4.z, ct4.w};
  float sl1 = 0.f;
  int np = 0;
#pragma unroll
  for (int e = 0; e < 4; ++e) {
    if (cts[e] > 0) {
      ++np;
      size_t li = (rowoff + (size_t)(base + e)) * 4;
      float4 ld = *(const float4*)(loc_data + li);
      float4 lt = *(const float4*)(loc_t + li);
      sl1 += smooth_l1(ld.x - lt.x) + smooth_l1(ld.y - lt.y) +
             smooth_l1(ld.z - lt.z) + smooth_l1(ld.w - lt.w);
    }
  }

  __shared__ float rf[256];
  __shared__ int ri[256];
  rf[threadIdx.x] = sl1;
  ri[threadIdx.x] = np;
  __syncthreads();
  for (int s = 128; s > 0; s >>= 1) {
    if (threadIdx.x < s) {
      rf[threadIdx.x] += rf[threadIdx.x + s];
      ri[threadIdx.x] += ri[threadIdx.x + s];
    }
    __syncthreads();
  }
  if (threadIdx.x == 0) {
    wsF[WS_LOCPART + b * LOC_BLOCKS_PER_ROW + chunk] = rf[0];
    atomicAdd(&wsI[WS_NUMPOS + b], ri[0]);  // integer atomic: deterministic
  }
}

// Pass 2: one workgroup per row. Async-copy the whole conf row into LDS
// (CDNA5 global_load_async_to_lds_b128, 256KB of the 320KB/WGP LDS), compute
// BCE in place, then 3-pass in-LDS radix select for the k-th largest negative
// BCE and the sum over selected negatives.
__global__ __launch_bounds__(1024) void conf_kernel(
    const float* __restrict__ conf_data, const int* __restrict__ conf_t,
    const int* __restrict__ wsI, float* __restrict__ wsF) {
  extern __shared__ float smem[];
  float* V = smem;                             // DDIM floats (row of bce_neg)
  unsigned* H = (unsigned*)(V + DDIM);         // 2048-bin histogram
  float* R = (float*)H;                        // reduction scratch (aliases H)
  unsigned* S = (unsigned*)(V + DDIM + 2048);  // small broadcast scratch

  const int b = blockIdx.x;
  const int tid = threadIdx.x;
  const size_t rowoff = (size_t)b * DDIM;
  const float* cp = conf_data + rowoff;
  const int* ct = conf_t + rowoff;

  // ---- Phase 0: async global -> LDS copy of the full row (16 b128 per wave) ----
#pragma unroll
  for (int it = 0; it < 16; ++it) {
    int fi = (it * 1024 + tid) * 4;  // 4 floats (16B) per lane per op
    unsigned lds_addr = (unsigned)(size_t)(&V[fi]);
    unsigned long long gaddr = (unsigned long long)(size_t)(cp + fi);
    asm volatile("global_load_async_to_lds_b128 %0, %1, off"
                 :: "v"(lds_addr), "v"(gaddr) : "memory");
  }
  asm volatile("s_wait_asynccnt 0" ::: "memory");
  __syncthreads();

  // ---- Phase 1: BCE in place; bce_neg = pos ? 0 : bce ----
  float posb = 0.f;
#pragma unroll 4
  for (int it = 0; it < DDIM / 1024; ++it) {
    int j = it * 1024 + tid;
    float x = V[j];
    int t = ct[j];
    float bce = fmaxf(x, 0.f) - ((t > 0) ? x : 0.f) + log1pf(expf(-fabsf(x)));
    bool pos = (t > 0);
    if (pos) posb += bce;
    V[j] = pos ? 0.f : bce;
  }
  // deterministic tree reduction of positive-bce sum
  R[tid] = posb;
  __syncthreads();
  for (int s = 512; s > 0; s >>= 1) {
    if (tid < s) R[tid] += R[tid + s];
    __syncthreads();
  }
  if (tid == 0) wsF[WS_POSBCE + b] = R[0];

  int npos = wsI[WS_NUMPOS + b];
  int k = npos * NEGPOS;
  if (k > DDIM) k = DDIM;
  if (k <= 0) {
    if (tid == 0) wsF[WS_NEGSEL + b] = 0.f;
    return;
  }
  __syncthreads();  // R (=H) reduction done before reusing H

  // ---- Phase 2: 3-pass radix select (bce >= 0 => uint order == float order) ----
  unsigned kk = (unsigned)k;
  unsigned t1, t2, t3;

  // pass 1: bits[31:21]
  for (int i = tid; i < 2048; i += 1024) H[i] = 0;
  __syncthreads();
#pragma unroll 4
  for (int it = 0; it < DDIM / 1024; ++it) {
    unsigned bits = __float_as_uint(V[it * 1024 + tid]);
    atomicAdd(&H[bits >> 21], 1u);
  }
  __syncthreads();
  if (tid == 0) {
    unsigned cum = 0, bin = 0, rem = kk;
    for (int i = 2047; i >= 0; --i) {
      unsigned c = H[i];
      if (cum + c >= kk) { bin = (unsigned)i; rem = kk - cum; break; }
      cum += c;
    }
    S[0] = bin; S[1] = rem;
  }
  __syncthreads();
  t1 = S[0]; kk = S[1];
  __syncthreads();

  // pass 2: bits[20:10]
  for (int i = tid; i < 2048; i += 1024) H[i] = 0;
  __syncthreads();
#pragma unroll 4
  for (int it = 0; it < DDIM / 1024; ++it) {
    unsigned bits = __float_as_uint(V[it * 1024 + tid]);
    if ((bits >> 21) == t1) atomicAdd(&H[(bits >> 10) & 0x7FFu], 1u);
  }
  __syncthreads();
  if (tid == 0) {
    unsigned cum = 0, bin = 0, rem = kk;
    for (int i = 2047; i >= 0; --i) {
      unsigned c = H[i];
      if (cum + c >= kk) { bin = (unsigned)i; rem = kk - cum; break; }
      cum += c;
    }
    S[0] = bin; S[1] = rem;
  }
  __syncthreads();
  t2 = S[0]; kk = S[1];
  __syncthreads();

  // pass 3: bits[9:0]
  for (int i = tid; i < 2048; i += 1024) H[i] = 0;
  __syncthreads();
  unsigned hi21 = (t1 << 11) | t2;
#pragma unroll 4
  for (int it = 0; it < DDIM / 1024; ++it) {
    unsigned bits = __float_as_uint(V[it * 1024 + tid]);
    if ((bits >> 10) == hi21) atomicAdd(&H[bits & 0x3FFu], 1u);
  }
  __syncthreads();
  if (tid == 0) {
    unsigned cum = 0, bin = 0;
    for (int i = 1023; i >= 0; --i) {
      unsigned c = H[i];
      if (cum + c >= kk) { bin = (unsigned)i; break; }
      cum += c;
    }
    S[0] = bin;
  }
  __syncthreads();
  t3 = S[0];
  unsigned T = (t1 << 21) | (t2 << 10) | t3;  // k-th largest value's bits
  __syncthreads();

  // ---- Phase 3: sum of selected negatives = sum(bce > T) + (k - cnt_gt) * T ----
  float sgt = 0.f;
  int cgt = 0;
#pragma unroll 4
  for (int it = 0; it < DDIM / 1024; ++it) {
    float v = V[it * 1024 + tid];
    unsigned bits = __float_as_uint(v);
    if (bits > T) { sgt += v; ++cgt; }
  }
  R[tid] = sgt;
  __syncthreads();
  for (int s = 512; s > 0; s >>= 1) {
    if (tid < s) R[tid] += R[tid + s];
    __syncthreads();
  }
  if (tid == 0) ((float*)S)[2] = R[0];
  __syncthreads();
  int* Ri = (int*)R;
  Ri[tid] = cgt;
  __syncthreads();
  for (int s = 512; s > 0; s >>= 1) {
    if (tid < s) Ri[tid] += Ri[tid + s];
    __syncthreads();
  }
  if (tid == 0) {
    float sumGT = ((float*)S)[2];
    int cntGT = Ri[0];
    float thr = __uint_as_float(T);
    wsF[WS_NEGSEL + b] = sumGT + (float)(k - cntGT) * thr;
  }
}

// Pass 3: deterministic combine of all partials into the two scalar losses.
__global__ __launch_bounds__(1024) void final_kernel(
    const int* __restrict__ wsI, const float* __restrict__ wsF, float* __restrict__ out) {
  __shared__ float R[1024];
  __shared__ float rowc[BDIM];
  __shared__ int rowp[BDIM];
  __shared__ int rowsel[BDIM];
  int tid = threadIdx.x;

  float s = 0.f;
#pragma unroll
  for (int e = 0; e < NPART / 1024; ++e) s += wsF[WS_LOCPART + e * 1024 + tid];
  R[tid] = s;
  if (tid < BDIM) {
    int np = wsI[WS_NUMPOS + tid];
    int nn = np * NEGPOS;
    if (nn > DDIM) nn = DDIM;
    rowp[tid] = np;
    rowsel[tid] = np + nn;
    rowc[tid] = wsF[WS_POSBCE + tid] + wsF[WS_NEGSEL + tid];
  }
  __syncthreads();
  for (int sft = 512; sft > 0; sft >>= 1) {
    if (tid < sft) R[tid] += R[tid + sft];
    __syncthreads();
  }
  if (tid == 0) {
    float sl1 = R[0];
    float csum = 0.f;
    long long tp = 0, ts = 0;
    for (int i = 0; i < BDIM; ++i) { csum += rowc[i]; tp += rowp[i]; ts += rowsel[i]; }
    float N = (float)tp;
    float loss_l = sl1 / (N * 4.0f);
    float loss_c = csum / (float)ts;
    out[0] = loss_l / N;
    out[1] = loss_c / N;
  }
}

extern "C" void kernel_launch(void* const* d_in, const int* in_sizes, int n_in,
                              void* d_out, int out_size, void* d_ws, size_t ws_size,
                              hipStream_t stream) {
  const float* loc_data = (const float*)d_in[0];
  const float* conf_data = (const float*)d_in[1];
  const float* loc_t = (const float*)d_in[2];
  const int* conf_t = (const int*)d_in[3];
  float* out = (float*)d_out;
  int* wsI = (int*)d_ws;
  float* wsF = (float*)d_ws;

  (void)in_sizes; (void)n_in; (void)out_size; (void)ws_size;

  // allow 264KB+ dynamic LDS (CDNA5: 320KB per WGP)
  hipFuncSetAttribute((const void*)conf_kernel,
                      hipFuncAttributeMaxDynamicSharedMemorySize, SMEM_BYTES);

  init_ws_kernel<<<1, 64, 0, stream>>>(wsI);
  loc_kernel<<<dim3(LOC_BLOCKS_PER_ROW, BDIM), 256, 0, stream>>>(
      loc_data, loc_t, conf_t, wsI, wsF);
  conf_kernel<<<BDIM, 1024, SMEM_BYTES, stream>>>(conf_data, conf_t, wsI, wsF);
  final_kernel<<<1, 1024, 0, stream>>>(wsI, wsF, out);
}